// MMDDistance_12884901888311
// MI455X (gfx1250) — compile-verified
//
#include <hip/hip_runtime.h>
#include <hip/hip_bf16.h>
#include <math.h>

typedef float v2f __attribute__((ext_vector_type(2)));
typedef float v8f __attribute__((ext_vector_type(8)));

#define NX 4096          // rows in x (and in y)
#define NT 8192          // total rows
#define DD 256           // feature dim

// Block geometry: 256 threads = 8 waves. Each wave computes a 16(M) x 64(N)
// strip (4 WMMA accumulators, shared A fragment). Block = 128(M) x 64(N).
#define WAVES 8
#define BLK_M 128
#define BLK_N 64
#define NSUB  4          // 16-wide N-subtiles per wave

// LDS B-panel: 64 rows x 256 k, padded row stride (floats).
// 260*4 = 1040 B row pitch: 16B-aligned (b128 stores OK), bank stride 4
// -> 16 lanes of a half-wave hit 16 distinct banks on ds_load_b64.
#define BST 260
#define SH_BYTES (BLK_N * BST * 4)

__device__ __forceinline__ const float* row_ptr(const float* __restrict__ x,
                                                const float* __restrict__ y,
                                                int i) {
  return (i < NX) ? (x + (size_t)i * DD) : (y + (size_t)(i - NX) * DD);
}

// ---------------- kernel 1: row squared norms ----------------
__global__ void row_norms_kernel(const float* __restrict__ x,
                                 const float* __restrict__ y,
                                 float* __restrict__ sq) {
  int i = blockIdx.x * blockDim.x + threadIdx.x;
  if (i >= NT) return;
  const float* p = row_ptr(x, y, i);
  float s = 0.f;
#pragma unroll 8
  for (int k = 0; k < DD; k += 4) {
    float4 v = *(const float4*)(p + k);
    s += v.x * v.x + v.y * v.y + v.z * v.z + v.w * v.w;
  }
  sq[i] = s;
}

// ---------------- unified GEMM pass ----------------
// PASS 0: accumulate sum of clamped L2 distances (for bandwidth).
// PASS 1: accumulate signed 5-scale Gaussian kernel sum (MMD numerator).
template <int PASS>
__global__ void gemm_pass_kernel(const float* __restrict__ x,
                                 const float* __restrict__ y,
                                 const float* __restrict__ sq,
                                 const double* __restrict__ bw_in,
                                 float* __restrict__ partial) {
  extern __shared__ float bsh[];            // [BLK_N][BST]

  const int lane = threadIdx.x & 31;
  const int wave = threadIdx.x >> 5;
  const int half = lane >> 4;               // K-pair select per ISA A/B layout
  const int l15  = lane & 15;

  const int tnBase = blockIdx.y * BLK_N;
  const int tm     = blockIdx.x * BLK_M + wave * 16;

  // ---- stage B panel (rows tnBase..tnBase+63, all K) into LDS ----
  for (int i = threadIdx.x; i < BLK_N * (DD / 4); i += 256) {
    const int r  = i >> 6;                  // 0..63  (DD/4 == 64)
    const int k4 = i & 63;                  // float4 index along K
    float4 v = *(const float4*)(row_ptr(x, y, tnBase + r) + k4 * 4);
    *(float4*)(&bsh[r * BST + k4 * 4]) = v; // ds_store_b128
  }
  __syncthreads();

  // ---- WMMA main loop: 1 global b64 (A) feeds 4 WMMAs per k-step ----
  const float* arow  = row_ptr(x, y, tm + l15) + 2 * half;
  const float* bbase = &bsh[l15 * BST + 2 * half];

  v8f acc[NSUB];
#pragma unroll
  for (int j = 0; j < NSUB; ++j)
    acc[j] = (v8f){0.f, 0.f, 0.f, 0.f, 0.f, 0.f, 0.f, 0.f};

#pragma unroll 8
  for (int k = 0; k < DD; k += 4) {
    v2f a = *(const v2f*)(arow + k);
#pragma unroll
    for (int j = 0; j < NSUB; ++j) {
      v2f b = *(const v2f*)(bbase + j * 16 * BST + k);  // ds_load_b64
      acc[j] = __builtin_amdgcn_wmma_f32_16x16x4_f32(
          /*neg_a=*/false, a, /*neg_b=*/false, b,
          /*c_mod=*/(short)0, acc[j], /*reuse_a=*/false, /*reuse_b=*/false);
    }
  }

  // ---- epilogue ----
  const float inv_bw = (PASS == 1) ? (float)(1.0 / bw_in[0]) : 0.f;
  float s = 0.f;
#pragma unroll
  for (int j = 0; j < NSUB; ++j) {
    const int   col  = tnBase + j * 16 + l15;
    const float sqc  = sq[col];
    const bool  cpos = (col < NX);
#pragma unroll
    for (int v = 0; v < 8; ++v) {
      const int row = tm + v + 8 * half;    // C/D layout: VGPR v -> M=v (+8 hi half)
      float l2 = fmaxf(sq[row] + sqc - 2.0f * acc[j][v], 0.f);
      if (PASS == 0) {
        s += l2;
      } else {
        float t = -l2 * inv_bw;             // exp(-L2/(bw*2^p)) = exp(t/2^p)
        float e = 0.f, f = t;
#pragma unroll
        for (int p = 0; p < 5; ++p) { e += __expf(f); f *= 0.5f; }
        const bool rpos = (row < NX);
        s += (rpos == cpos) ? e : -e;
      }
    }
  }

  // ---- deterministic block reduction ----
#pragma unroll
  for (int off = 16; off > 0; off >>= 1) s += __shfl_xor(s, off, 32);
  __shared__ float wsum[WAVES];
  if (lane == 0) wsum[wave] = s;
  __syncthreads();
  if (threadIdx.x == 0) {
    float t = 0.f;
#pragma unroll
    for (int w = 0; w < WAVES; ++w) t += wsum[w];
    partial[blockIdx.y * gridDim.x + blockIdx.x] = t;
  }
}

// ---------------- fixed-order reduce -> bandwidth ----------------
__global__ void reduce_bw_kernel(const float* __restrict__ partial, int n,
                                 double* __restrict__ bw_out) {
  __shared__ double sh[256];
  double s = 0.0;
  for (int i = threadIdx.x; i < n; i += 256) s += (double)partial[i];
  sh[threadIdx.x] = s;
  __syncthreads();
  for (int off = 128; off > 0; off >>= 1) {
    if ((int)threadIdx.x < off) sh[threadIdx.x] += sh[threadIdx.x + off];
    __syncthreads();
  }
  if (threadIdx.x == 0) {
    const double nn = (double)NT * (double)NT - (double)NT;
    bw_out[0] = (sh[0] / nn) * 0.25;        // / KERNEL_MUL^(KERNEL_NUM//2)
  }
}

// ---------------- fixed-order reduce -> output ----------------
__global__ void reduce_out_kernel(const float* __restrict__ partial, int n,
                                  float* __restrict__ out) {
  __shared__ double sh[256];
  double s = 0.0;
  for (int i = threadIdx.x; i < n; i += 256) s += (double)partial[i];
  sh[threadIdx.x] = s;
  __syncthreads();
  for (int off = 128; off > 0; off >>= 1) {
    if ((int)threadIdx.x < off) sh[threadIdx.x] += sh[threadIdx.x + off];
    __syncthreads();
  }
  if (threadIdx.x == 0) {
    out[0] = (float)(sh[0] / ((double)NX * (double)NX));
  }
}

extern "C" void kernel_launch(void* const* d_in, const int* in_sizes, int n_in,
                              void* d_out, int out_size, void* d_ws, size_t ws_size,
                              hipStream_t stream) {
  const float* x = (const float*)d_in[0];
  const float* y = (const float*)d_in[1];
  float* out = (float*)d_out;

  // Workspace layout (everything fully overwritten before read; no zeroing):
  char* ws = (char*)d_ws;
  double* bw       = (double*)(ws + 0);                  // 8 B
  float*  sq       = (float*)(ws + 256);                 // 8192 f32 = 32 KB
  float*  partialA = (float*)(ws + 256 + 32768 + 256);   // 8192 f32 = 32 KB
  float*  partialB = (float*)(ws + 256 + 32768 + 256 + 32768 + 256);

  const dim3 tileGrid(NT / BLK_M, NT / BLK_N);           // (64, 128) blocks
  const int  nPartial = (NT / BLK_M) * (NT / BLK_N);     // 8192 partials

  row_norms_kernel<<<NT / 256, 256, 0, stream>>>(x, y, sq);
  gemm_pass_kernel<0><<<tileGrid, 256, SH_BYTES, stream>>>(x, y, sq, bw, partialA);
  reduce_bw_kernel<<<1, 256, 0, stream>>>(partialA, nPartial, bw);
  gemm_pass_kernel<1><<<tileGrid, 256, SH_BYTES, stream>>>(x, y, sq, bw, partialB);
  reduce_out_kernel<<<1, 256, 0, stream>>>(partialB, nPartial, out);
}